// MinHasher_55310588838270
// MI455X (gfx1250) — compile-verified
//
#include <hip/hip_runtime.h>

#define MODP   0x7FFFFFFFu     // 2^31 - 1 (Mersenne prime)
#define KHASH  128
#define WAVES  8               // 8 wave32 per block = 256 threads
#define TILE   64              // ids staged per wave per iteration

static __device__ __forceinline__ unsigned umin32(unsigned x, unsigned y) {
    return x < y ? x : y;      // v_min_u32
}

// (a*x + b) mod P with a < P, x < 2^20, so v < 2^52.
// One Mersenne fold gives r < 2^31 + 2^21 < 2P, then umin(r, r-P) canonicalizes:
// if r < P, r-P wraps to a huge value; if P <= r < 2P, r-P is the answer.
static __device__ __forceinline__ unsigned hash_mod(unsigned a, unsigned x, unsigned b) {
    unsigned long long v = (unsigned long long)a * x + b;        // v_mad_u64_u32
    unsigned r = ((unsigned)v & MODP) + (unsigned)(v >> 31);     // fold 2^31 == 1 (mod P)
    return umin32(r, r - MODP);                                  // canonical [0, P)
}

__global__ __launch_bounds__(256)
void minhash_kernel(const int* __restrict__ ids,
                    const int* __restrict__ offsets,
                    const int* __restrict__ hash_a,
                    const int* __restrict__ hash_b,
                    float* __restrict__ out,
                    int num_sets, int nnz)
{
    __shared__ __align__(16) int tile[WAVES][TILE];   // 2 KB / block, one slice per wave

    const int wave = threadIdx.x >> 5;
    const int lane = threadIdx.x & 31;
    const int s    = blockIdx.x * WAVES + wave;
    if (s >= num_sets) return;                        // wave-uniform exit

    // Lane l owns hash functions l, l+32, l+64, l+96 (K = 128, wave32).
    const unsigned a0 = (unsigned)hash_a[lane      ], b0 = (unsigned)hash_b[lane      ];
    const unsigned a1 = (unsigned)hash_a[lane + 32 ], b1 = (unsigned)hash_b[lane + 32 ];
    const unsigned a2 = (unsigned)hash_a[lane + 64 ], b2 = (unsigned)hash_b[lane + 64 ];
    const unsigned a3 = (unsigned)hash_a[lane + 96 ], b3 = (unsigned)hash_b[lane + 96 ];

    const int start = offsets[s];
    const int end   = offsets[s + 1];

    // Reference identity for empty segments is P-1; h < P never exceeds it.
    unsigned m0 = MODP - 1, m1 = MODP - 1, m2 = MODP - 1, m3 = MODP - 1;

    // LDS byte addresses for this lane's two staging slots (low 32 bits of the
    // generic address of a __shared__ object are the LDS offset on gfx1250).
    const unsigned lds0 = (unsigned)(unsigned long long)(&tile[wave][lane]);
    const unsigned lds1 = lds0 + 32u * 4u;

    for (int t = start; t < end; t += TILE) {
        const int cnt = min(TILE, end - t);

        // Clamp (instead of predicating) so EXEC stays all-ones for async ops;
        // clamped garbage lands at positions >= cnt and is never read.
        const int gi0 = min(t + lane,      nnz - 1);
        const int gi1 = min(t + lane + 32, nnz - 1);
        const unsigned long long ga0 = (unsigned long long)(ids + gi0);
        const unsigned long long ga1 = (unsigned long long)(ids + gi1);

        // Drain pending ds_loads before overwriting the slice (multi-tile sets),
        // then async-stage 64 ids global -> LDS (ASYNCcnt path, CDNA5).
        asm volatile("s_wait_dscnt 0" ::: "memory");
        asm volatile("global_load_async_to_lds_b32 %0, %1, off"
                     :: "v"(lds0), "v"(ga0) : "memory");
        asm volatile("global_load_async_to_lds_b32 %0, %1, off"
                     :: "v"(lds1), "v"(ga1) : "memory");
        asm volatile("s_wait_asynccnt 0" ::: "memory");

        #define UPD(xx) { const unsigned ux = (unsigned)(xx);      \
            m0 = umin32(m0, hash_mod(a0, ux, b0));                 \
            m1 = umin32(m1, hash_mod(a1, ux, b1));                 \
            m2 = umin32(m2, hash_mod(a2, ux, b2));                 \
            m3 = umin32(m3, hash_mod(a3, ux, b3)); }

        const int nv = cnt & ~3;
        for (int e = 0; e < nv; e += 4) {
            // 16B-aligned broadcast read: one ds_load_b128 feeds 16 hash evals.
            const int4 x4 = *(const int4*)&tile[wave][e];
            UPD(x4.x); UPD(x4.y); UPD(x4.z); UPD(x4.w);
        }
        for (int e = nv; e < cnt; ++e) {
            UPD(tile[wave][e]);
        }
        #undef UPD
    }

    // Coalesced: 4 x 128B stores per wave.
    float* o = out + (size_t)s * KHASH;
    o[lane      ] = (float)m0;
    o[lane + 32 ] = (float)m1;
    o[lane + 64 ] = (float)m2;
    o[lane + 96 ] = (float)m3;
}

extern "C" void kernel_launch(void* const* d_in, const int* in_sizes, int n_in,
                              void* d_out, int out_size, void* d_ws, size_t ws_size,
                              hipStream_t stream) {
    const int* ids     = (const int*)d_in[0];
    const int* offsets = (const int*)d_in[1];
    const int* a       = (const int*)d_in[2];
    const int* b       = (const int*)d_in[3];
    float* out         = (float*)d_out;

    const int nnz      = in_sizes[0];
    const int num_sets = in_sizes[1] - 1;   // offsets has S+1 entries

    const int blocks = (num_sets + WAVES - 1) / WAVES;
    minhash_kernel<<<dim3(blocks), dim3(256), 0, stream>>>(
        ids, offsets, a, b, out, num_sets, nnz);
}